// GCN_LP_33380485824613
// MI455X (gfx1250) — compile-verified
//
#include <hip/hip_runtime.h>

typedef float v2f __attribute__((ext_vector_type(2)));
typedef float v8f __attribute__((ext_vector_type(8)));

#define LDS_STRIDE 132  // 128 + 4 pad: breaks 64-bank conflict for stride-128 accesses

// ---------------- degree / normalization ----------------
__global__ void deg_init_kernel(float* deg, int n) {
  int i = blockIdx.x * blockDim.x + threadIdx.x;
  if (i < n) deg[i] = 1.0f;  // self-loop
}

__global__ void deg_count_kernel(const int* __restrict__ dst, float* __restrict__ deg, int nE) {
  int e = blockIdx.x * blockDim.x + threadIdx.x;
  if (e < nE) unsafeAtomicAdd(&deg[dst[e]], 1.0f);
}

__global__ void deg_rsqrt_kernel(float* deg, int n) {
  int i = blockIdx.x * blockDim.x + threadIdx.x;
  if (i < n) deg[i] = rsqrtf(deg[i]);  // deg >= 1 always
}

// ---------------- fused gather + GEMM (WMMA f32 16x16x4) + dinv scale ----------------
// hs[r,:] = (X[gidx ? gidx[r] : r, :] @ W) * dinv[r];  agg[r,:] = hs[r,:] (self-loop preinit)
// X: rows of K=128 f32.  W: 128 x N row-major.  256 threads = 8 waves; 64 rows per block.
__global__ void gemm_gather_scale_kernel(const float* __restrict__ X,
                                         const int* __restrict__ gidx,
                                         const float* __restrict__ W,
                                         const float* __restrict__ dinv,
                                         float* __restrict__ hs,
                                         float* __restrict__ agg,
                                         int n_rows, int N) {
  extern __shared__ float lds[];
  float* lA = lds;                    // 64 x LDS_STRIDE
  float* lW = lds + 64 * LDS_STRIDE;  // N x LDS_STRIDE  (lW[n][k] = W[k][n])
  const int tid = threadIdx.x;
  const int row0 = blockIdx.x * 64;

  // Stage gathered A-tile (64 x 128) as float4, zero-padded past n_rows.
  const float4* X4 = (const float4*)X;
  for (int i = tid; i < 64 * 32; i += 256) {
    int r = i >> 5;
    int c4 = i & 31;
    int gr = row0 + r;
    float4 v = make_float4(0.f, 0.f, 0.f, 0.f);
    if (gr < n_rows) {
      int sr = gidx ? gidx[gr] : gr;
      v = X4[(size_t)sr * 32 + c4];
    }
    *(float4*)&lA[r * LDS_STRIDE + (c4 << 2)] = v;
  }
  // Stage W transposed so B fragments are contiguous-in-K per lane.
  for (int i = tid; i < 128 * N; i += 256) {
    int k = i / N;
    int n = i - k * N;
    lW[n * LDS_STRIDE + k] = W[i];
  }
  __syncthreads();

  const int wave = tid >> 5;
  const int lane = tid & 31;
  const int nct = N >> 4;             // 16-col tiles
  const int col = lane & 15;          // A: row M; B: col N; C: col N
  const int koff = (lane >> 4) << 1;  // lanes 16-31 hold K=k0+2,k0+3 (ISA 7.12.2)

  // 4 row-tiles x nct col-tiles; trip count uniform across waves (32 or 16, /8).
  for (int t = wave; t < 4 * nct; t += 8) {
    int rt = t / nct;
    int ct = t - rt * nct;
    int mbase = rt << 4;
    int nbase = ct << 4;
    const float* aRow = &lA[(mbase + col) * LDS_STRIDE + koff];
    const float* bRow = &lW[(nbase + col) * LDS_STRIDE + koff];
    v8f c = {0.f, 0.f, 0.f, 0.f, 0.f, 0.f, 0.f, 0.f};
#pragma unroll
    for (int k0 = 0; k0 < 128; k0 += 4) {
      v2f a = *(const v2f*)(aRow + k0);  // ds_load_b64 (8B aligned: stride & koff even)
      v2f b = *(const v2f*)(bRow + k0);
      c = __builtin_amdgcn_wmma_f32_16x16x4_f32(false, a, false, b, (short)0, c, false, false);
    }
    // C layout: VGPR i -> row M=i (lanes 0-15) / M=i+8 (lanes 16-31), col = lane&15.
    int madd = (lane < 16) ? 0 : 8;
#pragma unroll
    for (int i = 0; i < 8; i++) {
      int gr = row0 + mbase + i + madd;
      if (gr < n_rows) {
        float v = c[i] * dinv[gr];
        size_t o = (size_t)gr * N + nbase + col;
        hs[o] = v;
        agg[o] = v;
      }
    }
  }
}

// ---------------- edge scatter: agg[dst,:] += hs[src,:] ----------------
__global__ void scatter_kernel(const int* __restrict__ src, const int* __restrict__ dst,
                               const float* __restrict__ hs, float* __restrict__ agg,
                               int nE, int N) {
  int g = blockIdx.x * blockDim.x + threadIdx.x;
  int e = g >> 5;
  int lane = g & 31;
  if (e >= nE) return;
  const float* hp = hs + (size_t)src[e] * N;
  float* ap = agg + (size_t)dst[e] * N;
  for (int f = lane; f < N; f += 32)
    unsafeAtomicAdd(&ap[f], hp[f]);  // hardware global_atomic_add_f32
}

// ---------------- out = [relu](dinv[r]*agg + bias[f]) ----------------
__global__ void combine_kernel(const float* __restrict__ agg, const float* __restrict__ dinv,
                               const float* __restrict__ bias, float* __restrict__ out,
                               int n_rows, int N, int do_relu) {
  int i = blockIdx.x * blockDim.x + threadIdx.x;
  if (i >= n_rows * N) return;
  int r = i / N;
  int f = i - r * N;
  float v = dinv[r] * agg[i] + bias[f];
  if (do_relu) v = fmaxf(v, 0.f);
  out[i] = v;
}

// ---------------- link prediction: out[e] = dot(z[s], z[d]) over 64 dims ----------------
__global__ void edgedot_kernel(const int* __restrict__ es, const int* __restrict__ ed,
                               const float* __restrict__ z, float* __restrict__ out, int nL) {
  int g = blockIdx.x * blockDim.x + threadIdx.x;
  int e = g >> 5;
  int lane = g & 31;
  if (e >= nL) return;
  const float* a = z + (size_t)es[e] * 64;
  const float* b = z + (size_t)ed[e] * 64;
  float p = a[lane] * b[lane] + a[lane + 32] * b[lane + 32];
#pragma unroll
  for (int off = 16; off > 0; off >>= 1) p += __shfl_xor(p, off, 32);
  if (lane == 0) out[e] = p;
}

extern "C" void kernel_launch(void* const* d_in, const int* in_sizes, int n_in,
                              void* d_out, int out_size, void* d_ws, size_t ws_size,
                              hipStream_t stream) {
  const int*   x_idx = (const int*)d_in[0];    // (100000,1) int32
  const int*   eidx  = (const int*)d_in[1];    // (2, nE)
  const int*   lidx  = (const int*)d_in[2];    // (2, nL)
  const float* emb   = (const float*)d_in[3];  // 10000 x 128
  const float* W1    = (const float*)d_in[4];  // 128 x 128
  const float* b1    = (const float*)d_in[5];  // 128
  const float* W2    = (const float*)d_in[6];  // 128 x 64
  const float* b2    = (const float*)d_in[7];  // 64
  float* out = (float*)d_out;

  const int nN = in_sizes[0];
  const int nE = in_sizes[1] / 2;
  const int nL = in_sizes[2] / 2;
  const int H = 128, O = 64;

  // workspace layout (f32): dinv | hs1(nN*128) | agg1(nN*128) | hs2(nN*64) | agg2(nN*64)
  float* dinv = (float*)d_ws;
  float* hs1  = dinv + ((nN + 63) & ~63);
  float* agg1 = hs1 + (size_t)nN * H;
  float* hs2  = agg1 + (size_t)nN * H;
  float* agg2 = hs2 + (size_t)nN * O;

  const int* esrc = eidx;       // edge_index[0]
  const int* edst = eidx + nE;  // edge_index[1]

  // deg = 1 + in-degree; dinv = rsqrt(deg)
  deg_init_kernel<<<(nN + 255) / 256, 256, 0, stream>>>(dinv, nN);
  deg_count_kernel<<<(nE + 255) / 256, 256, 0, stream>>>(edst, dinv, nE);
  deg_rsqrt_kernel<<<(nN + 255) / 256, 256, 0, stream>>>(dinv, nN);

  // Layer 1: hs1 = (emb[x_idx] @ W1) * dinv; agg1 preinit = hs1 (self-loop)
  size_t shmem1 = (size_t)(64 * LDS_STRIDE + 128 * LDS_STRIDE) * sizeof(float);
  gemm_gather_scale_kernel<<<(nN + 63) / 64, 256, shmem1, stream>>>(
      emb, x_idx, W1, dinv, hs1, agg1, nN, H);
  scatter_kernel<<<(unsigned)(((size_t)nE * 32 + 255) / 256), 256, 0, stream>>>(
      esrc, edst, hs1, agg1, nE, H);
  combine_kernel<<<(unsigned)(((size_t)nN * H + 255) / 256), 256, 0, stream>>>(
      agg1, dinv, b1, hs1, nN, H, /*relu=*/1);  // x1 overwrites hs1 buffer

  // Layer 2: hs2 = (x1 @ W2) * dinv; agg2 preinit = hs2
  size_t shmem2 = (size_t)(64 * LDS_STRIDE + 64 * LDS_STRIDE) * sizeof(float);
  gemm_gather_scale_kernel<<<(nN + 63) / 64, 256, shmem2, stream>>>(
      hs1, nullptr, W2, dinv, hs2, agg2, nN, O);
  scatter_kernel<<<(unsigned)(((size_t)nE * 32 + 255) / 256), 256, 0, stream>>>(
      esrc, edst, hs2, agg2, nE, O);
  combine_kernel<<<(unsigned)(((size_t)nN * O + 255) / 256), 256, 0, stream>>>(
      agg2, dinv, b2, hs2, nN, O, /*relu=*/0);  // z overwrites hs2 buffer

  // out[e] = dot(z[lidx0[e]], z[lidx1[e]])
  edgedot_kernel<<<(unsigned)(((size_t)nL * 32 + 255) / 256), 256, 0, stream>>>(
      lidx, lidx + nL, hs2, out, nL);
}